// JonesModel_23390391894596
// MI455X (gfx1250) — compile-verified
//
#include <hip/hip_runtime.h>

// Problem constants (match reference)
#define NPOLP   4        // 2x2 pol pairs flattened
#define NANT    64
#define NBL     2016
#define NTIME   64
#define NFREQ   256
#define FTILE   64                   // freqs handled per block
#define ANT_ROW (FTILE + 4)          // padded floats per (polpair, ant) row -> 68
#define LDS_FLOATS (NPOLP * NANT * ANT_ROW)   // 4*64*68 = 17408 floats = 69632 B

typedef float v4f __attribute__((ext_vector_type(4)));

__global__ __launch_bounds__(256) void jones_apply_kernel(
    const float* __restrict__ Vm,      // [2][2][NBL][NTIME][NFREQ]
    const float* __restrict__ jones,   // [2][2][NANT][NTIME][NFREQ]
    const int*   __restrict__ ant1,    // [NBL]
    const int*   __restrict__ ant2,    // [NBL]
    float*       __restrict__ out)     // [2][2][NBL][NTIME][NFREQ]
{
    __shared__ float sj[LDS_FLOATS];

    const int tid = threadIdx.x;           // 0..255
    const int t   = blockIdx.x >> 2;       // 0..63
    const int ft  = blockIdx.x & 3;        // freq tile 0..3
    const int f0  = ft * FTILE;

    const size_t site   = (size_t)NTIME * NFREQ;   // 16384
    const size_t polstr = (size_t)NBL * site;      // pol-pair stride in Vm/out

    // ------------------------------------------------------------------
    // Stage jones[:, :, all 64 ants, t, f0..f0+63] into LDS with the
    // CDNA5 async copy engine: 4096 b128 transfers, 16 per thread.
    // ------------------------------------------------------------------
    #pragma unroll
    for (int k = 0; k < 16; ++k) {
        const int l   = tid + 256 * k;     // 0..4095
        const int f4  = l & 15;            // float4 column in tile
        const int ant = (l >> 4) & 63;
        const int pp  = l >> 10;           // 0..3 pol pair
        const float* g = jones + ((size_t)pp * NANT + ant) * site
                               + (size_t)t * NFREQ + f0 + 4 * f4;
        const unsigned ldsaddr =
            (unsigned)(size_t)&sj[(pp * NANT + ant) * ANT_ROW + 4 * f4];
        asm volatile("global_load_async_to_lds_b128 %0, %1, off"
                     :: "v"(ldsaddr), "v"((unsigned long long)g)
                     : "memory");
    }
#if __has_builtin(__builtin_amdgcn_s_wait_asynccnt)
    __builtin_amdgcn_s_wait_asynccnt(0);
#else
    asm volatile("s_wait_asynccnt 0x0" ::: "memory");
#endif
    __syncthreads();

    // ------------------------------------------------------------------
    // Main stream: 16 lanes per baseline cover 64 freqs as float4.
    // Lanes 0..15 -> contiguous 256B of the freq axis (coalesced).
    // ------------------------------------------------------------------
    const int fvec  = tid & 15;            // float4 column
    const int blsub = tid >> 4;            // baseline within group of 16
    const int fcol  = f0 + 4 * fvec;
    const int ppstr = NANT * ANT_ROW;      // pol-pair stride inside LDS tile

    for (int g = blockIdx.y; g < NBL / 16; g += gridDim.y) {
        const int n  = g * 16 + blsub;
        const int a1 = ant1[n];
        const int a2 = ant2[n];

        const float* sj1 = &sj[a1 * ANT_ROW + 4 * fvec];
        const float* sj2 = &sj[a2 * ANT_ROW + 4 * fvec];

        v4f j1[2][2], j2[2][2], V[2][2];
        #pragma unroll
        for (int a = 0; a < 2; ++a)
            #pragma unroll
            for (int b = 0; b < 2; ++b) {
                j1[a][b] = *(const v4f*)(sj1 + (a * 2 + b) * ppstr);
                j2[a][b] = *(const v4f*)(sj2 + (a * 2 + b) * ppstr);
            }

        // Streaming V_m load with non-temporal hint (don't pollute L2).
        const float* vg = Vm + (size_t)n * site + (size_t)t * NFREQ + fcol;
        #pragma unroll
        for (int b = 0; b < 2; ++b)
            #pragma unroll
            for (int c = 0; c < 2; ++c)
                V[b][c] = __builtin_nontemporal_load(
                              (const v4f*)(vg + (size_t)(b * 2 + c) * polstr));

        // W = J1 * V   (2x2 @ 2x2, elementwise over 4 freqs)
        v4f W[2][2];
        #pragma unroll
        for (int a = 0; a < 2; ++a)
            #pragma unroll
            for (int c = 0; c < 2; ++c)
                W[a][c] = j1[a][0] * V[0][c] + j1[a][1] * V[1][c];

        // out[a][d] = sum_c W[a][c] * j2[d][c]   (J2 conj == J2 for real)
        float* og = out + (size_t)n * site + (size_t)t * NFREQ + fcol;
        #pragma unroll
        for (int a = 0; a < 2; ++a)
            #pragma unroll
            for (int d = 0; d < 2; ++d) {
                v4f r = W[a][0] * j2[d][0] + W[a][1] * j2[d][1];
                __builtin_nontemporal_store(
                    r, (v4f*)(og + (size_t)(a * 2 + d) * polstr));
            }
    }
}

extern "C" void kernel_launch(void* const* d_in, const int* in_sizes, int n_in,
                              void* d_out, int out_size, void* d_ws, size_t ws_size,
                              hipStream_t stream) {
    const float* Vm    = (const float*)d_in[0];
    const float* jones = (const float*)d_in[1];
    const int*   ant1  = (const int*)d_in[2];
    const int*   ant2  = (const int*)d_in[3];
    float*       out   = (float*)d_out;

    // grid.x: 64 time steps x 4 freq tiles; grid.y strides the 126 groups
    // of 16 baselines (9 divides 126 evenly -> 14 iterations per block).
    dim3 grid(NTIME * (NFREQ / FTILE), 9);
    jones_apply_kernel<<<grid, 256, 0, stream>>>(Vm, jones, ant1, ant2, out);
}